// RUNG_41961830482001
// MI455X (gfx1250) — compile-verified
//
#include <hip/hip_runtime.h>
#include <math.h>

// ---------------- problem constants (match reference) ----------------
constexpr int   NN       = 4096;
constexpr int   IN_DIM   = 512;
constexpr int   HID      = 64;
constexpr int   OUT_DIM  = 16;
constexpr int   PROP_STEP = 10;
constexpr float GAMMA_F  = 3.0f;
constexpr int   CAP      = 256;   // max neighbors kept per row (mean ~65, safe)

typedef __attribute__((ext_vector_type(2))) float v2f;
typedef __attribute__((ext_vector_type(4))) float v4f;
typedef __attribute__((ext_vector_type(8))) float v8f;

// ---------------------------------------------------------------------
// Kernel 1: MLP  F0 = relu(F@W1 + b1) @ W2 + b2
// One wave (32 lanes) per 16-row tile; V_WMMA_F32_16X16X4_F32.
// A frag (16x4 f32): lanes 0-15 -> row m=lane, K = k+0,k+1 ; lanes 16-31 -> K = k+2,k+3
// B frag (4x16 f32): lane n = lane&15; low lanes K = k+0,k+1 ; high lanes K = k+2,k+3
// C/D (16x16 f32): elem r of v8f = row (r + 8*(lane>>4)), col = lane&15
// ---------------------------------------------------------------------
__global__ __launch_bounds__(32)
void mlp_kernel(const float* __restrict__ F,
                const float* __restrict__ W1, const float* __restrict__ b1,
                const float* __restrict__ W2, const float* __restrict__ b2,
                float* __restrict__ F0)
{
    __shared__ float Ht[16][HID];            // 16 x 64 f32 = 4 KB
    const int lane = threadIdx.x;
    const int half = lane >> 4;              // 0: K+0/1, 1: K+2/3
    const int l16  = lane & 15;
    const int row0 = blockIdx.x * 16;
    const int m    = l16;                    // A-fragment row for this lane

    const float* Frow = F + (size_t)(row0 + m) * IN_DIM + 2 * half;

    // ---- H = relu(F@W1 + b1), 16 rows x 64 cols, in 4 col-tiles of 16 ----
    for (int nt = 0; nt < HID / 16; ++nt) {
        const int ncol = nt * 16 + l16;
        v8f acc = {};
        for (int k = 0; k < IN_DIM; k += 4) {
            v2f a, b;
            a.x = Frow[k + 0];
            a.y = Frow[k + 1];
            b.x = W1[(size_t)(k + 2 * half + 0) * HID + ncol];
            b.y = W1[(size_t)(k + 2 * half + 1) * HID + ncol];
            acc = __builtin_amdgcn_wmma_f32_16x16x4_f32(
                      false, a, false, b, (short)0, acc, false, false);
        }
        const float bias = b1[ncol];
#pragma unroll
        for (int r = 0; r < 8; ++r) {
            float v = acc[r] + bias;
            Ht[r + 8 * half][ncol] = v > 0.0f ? v : 0.0f;
        }
    }
    __syncthreads();

    // ---- F0 tile = H @ W2 + b2, 16x16, K = 64 ----
    v8f acc2 = {};
    for (int k = 0; k < HID; k += 4) {
        v2f a, b;
        a.x = Ht[m][k + 2 * half + 0];
        a.y = Ht[m][k + 2 * half + 1];
        b.x = W2[(size_t)(k + 2 * half + 0) * OUT_DIM + l16];
        b.y = W2[(size_t)(k + 2 * half + 1) * OUT_DIM + l16];
        acc2 = __builtin_amdgcn_wmma_f32_16x16x4_f32(
                   false, a, false, b, (short)0, acc2, false, false);
    }
    const float bias2 = b2[l16];
#pragma unroll
    for (int r = 0; r < 8; ++r) {
        F0[(size_t)(row0 + r + 8 * half) * OUT_DIM + l16] = acc2[r] + bias2;
    }
}

// ---------------------------------------------------------------------
// Kernel 2: fused CSR build + degrees, single pass over A (the dominant
// 64 MB stream). A is exactly {0,1} with zero diagonal, so the row sum
// equals the nonzero count: D_i = count + 1 (exact in fp32).
// One wave per row; each lane loads float4 (128 cols per step, 4 ballots).
// Neighbor lists are emitted per 128-col step ordered by col%4, then column (deterministic).
// ---------------------------------------------------------------------
__global__ __launch_bounds__(32)
void csr_deg_kernel(const float* __restrict__ A,
                    int* __restrict__ deg, int* __restrict__ idx,
                    float* __restrict__ D, float* __restrict__ RS)
{
    const int i    = blockIdx.x;
    const int lane = threadIdx.x;
    const v4f* row4 = (const v4f*)(A + (size_t)i * NN);
    int* myidx = idx + (size_t)i * CAP;

    int count = 0;
    for (int j0 = 0; j0 < NN; j0 += 128) {
        const v4f v = row4[(j0 >> 2) + lane];
        const int jb = j0 + lane * 4;
#pragma unroll
        for (int q = 0; q < 4; ++q) {
            const int  j = jb + q;
            const bool p = (v[q] != 0.0f) && (j != i);
            const unsigned mask = (unsigned)__ballot(p);     // wave32
            const int pre = __popc(mask & ((lane == 0) ? 0u : ((1u << lane) - 1u)));
            if (p) {
                const int pos = count + pre;
                if (pos < CAP) myidx[pos] = j;
            }
            count += __popc(mask);
        }
    }
    if (lane == 0) {
        deg[i] = count < CAP ? count : CAP;
        const float d = (float)count + 1.0f;   // + self-loop; exact
        D[i]  = d;
        RS[i] = 1.0f / sqrtf(d);
    }
}

// ---------------------------------------------------------------------
// Kernel 3: one QN-IRLS propagation step. One wave per node.
// half-wave h handles neighbors e = h, h+2, ... ; lane channel c = lane&15.
// ---------------------------------------------------------------------
__global__ __launch_bounds__(32)
void prop_kernel(const float* __restrict__ Fin, const float* __restrict__ F0,
                 const float* __restrict__ D,   const float* __restrict__ RS,
                 const int* __restrict__ deg,   const int* __restrict__ idx,
                 float* __restrict__ Fout, float lam)
{
    const int i    = blockIdx.x;
    const int lane = threadIdx.x;
    const int half = lane >> 4;
    const int c    = lane & 15;

    const float rsi = RS[i];
    const float Xi  = Fin[(size_t)i * OUT_DIM + c] * rsi;   // X = Fc / sqrt(D)

    float acc  = 0.0f;   // sum_j w_ij * X_j[c]
    float sumw = 0.0f;   // sum_j w_ij  (replicated across the 16 lanes of a half)
    const int di = deg[i];

    for (int e = half; e < di; e += 2) {
        const int   j  = idx[(size_t)i * CAP + e];
        const float Xj = Fin[(size_t)j * OUT_DIM + c] * RS[j];
        float d = Xi - Xj;
        float z = d * d;
        z += __shfl_xor(z, 1, 32);           // masks < 16: stay inside half-wave
        z += __shfl_xor(z, 2, 32);
        z += __shfl_xor(z, 4, 32);
        z += __shfl_xor(z, 8, 32);
        const float y = sqrtf(z);
        float w = (y < GAMMA_F) ? (1.0f - y / GAMMA_F) / (2.0f * y) : 0.0f;
        if (w != w) w = 1.0f;                // reference NaN patch
        sumw += w;
        acc  += w * Xj;
    }
    // combine the two half-waves
    acc  += __shfl_xor(acc, 16, 32);
    sumw += __shfl_xor(sumw, 16, 32);

    const float Q   = sumw / D[i] + lam;     // Q_hat_i
    const float out = (rsi * acc) / Q + (lam * F0[(size_t)i * OUT_DIM + c]) / Q;
    if (half == 0) Fout[(size_t)i * OUT_DIM + c] = out;
}

// ---------------------------------------------------------------------
extern "C" void kernel_launch(void* const* d_in, const int* in_sizes, int n_in,
                              void* d_out, int out_size, void* d_ws, size_t ws_size,
                              hipStream_t stream)
{
    const float* A  = (const float*)d_in[0];
    const float* F  = (const float*)d_in[1];
    const float* W1 = (const float*)d_in[2];
    const float* b1 = (const float*)d_in[3];
    const float* W2 = (const float*)d_in[4];
    const float* b2 = (const float*)d_in[5];
    float* out = (float*)d_out;

    (void)in_sizes; (void)n_in; (void)out_size; (void)ws_size;

    // workspace layout (~4.8 MB)
    char*  ws  = (char*)d_ws;
    float* F0  = (float*)ws;  ws += (size_t)NN * OUT_DIM * sizeof(float);
    float* FcA = (float*)ws;  ws += (size_t)NN * OUT_DIM * sizeof(float);
    float* FcB = (float*)ws;  ws += (size_t)NN * OUT_DIM * sizeof(float);
    float* D   = (float*)ws;  ws += (size_t)NN * sizeof(float);
    float* RS  = (float*)ws;  ws += (size_t)NN * sizeof(float);
    int*   deg = (int*)ws;    ws += (size_t)NN * sizeof(int);
    int*   idx = (int*)ws;    ws += (size_t)NN * CAP * sizeof(int);

    const float lam = (float)(1.0 / 0.9 - 1.0);   // LAM = 1/LAM_HAT - 1

    mlp_kernel<<<NN / 16, 32, 0, stream>>>(F, W1, b1, W2, b2, F0);
    csr_deg_kernel<<<NN, 32, 0, stream>>>(A, deg, idx, D, RS);

    const float* cur = F0;
    for (int it = 0; it < PROP_STEP; ++it) {
        float* nxt = (it == PROP_STEP - 1) ? out : ((it & 1) ? FcB : FcA);
        prop_kernel<<<NN, 32, 0, stream>>>(cur, F0, D, RS, deg, idx, nxt, lam);
        cur = nxt;
    }
}